// MultiPositiveRerankerLoss_64209761075782
// MI455X (gfx1250) — compile-verified
//
#include <hip/hip_runtime.h>
#include <hip/hip_bf16.h>

typedef __attribute__((ext_vector_type(2))) float v2f;
typedef __attribute__((ext_vector_type(8))) float v8f;

#define MARGIN 0.1f
#define BATCH 512
#define NPOS 4
#define NNEG 64
#define DIM 1024
// scores layout in d_ws: [b][80] floats; tile0 cols0-3 = pos, tiles1-4 = neg
#define SCORE_STRIDE 80

// ---------------------------------------------------------------------------
// Stage 1: batched dot products via V_WMMA_F32_16X16X4_F32.
// grid = (5 tiles, 512 batches), block = 128 threads (4 waves).
// Each wave: 16 rows (B-matrix columns) x 256 d-values (64 WMMAs), query chunk
// replicated across the 16 A-rows so C[:,j] = dot(q, row_j) replicated over M.
// ---------------------------------------------------------------------------
__global__ void mprl_scores_kernel(const float* __restrict__ q,
                                   const float* __restrict__ pos,
                                   const float* __restrict__ neg,
                                   float* __restrict__ scores) {
  const int t    = blockIdx.x;            // tile: 0 = pos, 1..4 = neg quarters
  const int b    = blockIdx.y;            // batch
  const int tid  = threadIdx.x;
  const int w    = tid >> 5;              // wave 0..3 -> d-quarter
  const int lane = tid & 31;
  const int col  = lane & 15;             // B-matrix column = row-in-tile
  const int half = lane >> 4;             // K-half select (K0/1 vs K2/3)

  const float* rowbase;
  if (t == 0) {
    const int r = col < (NPOS - 1) ? col : (NPOS - 1);   // clamp; cols 4..15 dup row 3
    rowbase = pos + ((size_t)b * NPOS + r) * DIM;
  } else {
    const int r = (t - 1) * 16 + col;
    rowbase = neg + ((size_t)b * NNEG + r) * DIM;
  }
  const float* qbase = q + (size_t)b * DIM;

  v8f c = {};
  const int d0 = w * (DIM / 4) + 2 * half;   // this wave's d-quarter + K-half

#pragma unroll 8
  for (int k = 0; k < DIM / 16; ++k) {       // 64 WMMAs, step K=4
    const int d = d0 + 4 * k;
    // A fragment: q[d..d+1] (lanes 0-15) / q[d+2..d+3] (lanes 16-31),
    // identical for all 16 M-rows (broadcast load, L0-coalesced).
    v2f a = *(const v2f*)(qbase + d);
    // B fragment: 16 rows streamed once -> non-temporal.
    v2f bm = __builtin_nontemporal_load((const v2f*)(rowbase + d));
    // D = A(16x4) x B(4x16) + C : K-reduction in the matrix pipe.
    c = __builtin_amdgcn_wmma_f32_16x16x4_f32(
        /*neg_a=*/false, a, /*neg_b=*/false, bm,
        /*c_mod=*/(short)0, c, /*reuse_a=*/false, /*reuse_b=*/false);
  }

  // C VGPR0 on lane L = score of column L%16 (M replicated). Reduce the 4
  // d-quarter partials across waves through LDS.
  __shared__ float red[4][16];
  if (lane < 16) red[w][lane] = c[0];
  __syncthreads();
  if (tid < 16) {
    const float s = red[0][tid] + red[1][tid] + red[2][tid] + red[3][tid];
    scores[(size_t)b * SCORE_STRIDE + t * 16 + tid] = s;
  }
}

// ---------------------------------------------------------------------------
// Stage 2: scalar loss. One 64-thread block per batch, atomicAdd partials.
// ---------------------------------------------------------------------------
__global__ void mprl_zero_kernel(float* __restrict__ out) { out[0] = 0.0f; }

__global__ void mprl_loss_kernel(const float* __restrict__ scores,
                                 float* __restrict__ out) {
  const int b = blockIdx.x;
  const int j = threadIdx.x;               // 0..63 -> neg index
  const float* s = scores + (size_t)b * SCORE_STRIDE;

  __shared__ float pos_s[NPOS];
  __shared__ float red[NNEG];
  if (j < NPOS) pos_s[j] = s[j];           // tile 0, rows 0..3
  __syncthreads();

  const float nj = s[16 + j];              // tiles 1..4 are contiguous at +16
  float acc = 0.0f;
#pragma unroll
  for (int i = 0; i < NPOS; ++i) {
    const float v = MARGIN - pos_s[i] + nj;
    acc += v > 0.0f ? v : 0.0f;
  }
  red[j] = acc;
  __syncthreads();
#pragma unroll
  for (int o = 32; o > 0; o >>= 1) {
    if (j < o) red[j] += red[j + o];
    __syncthreads();
  }
  if (j == 0)
    atomicAdd(out, red[0] * (1.0f / (float)(BATCH * NPOS * NNEG)));
}

// ---------------------------------------------------------------------------
extern "C" void kernel_launch(void* const* d_in, const int* in_sizes, int n_in,
                              void* d_out, int out_size, void* d_ws, size_t ws_size,
                              hipStream_t stream) {
  const float* q   = (const float*)d_in[0];   // [512, 1024]
  const float* pos = (const float*)d_in[1];   // [512, 4, 1024]
  const float* neg = (const float*)d_in[2];   // [512, 64, 1024]
  float* out    = (float*)d_out;              // scalar f32
  float* scores = (float*)d_ws;               // 512*80 floats = 160 KB scratch

  mprl_zero_kernel<<<1, 1, 0, stream>>>(out);
  mprl_scores_kernel<<<dim3(5, BATCH), 128, 0, stream>>>(q, pos, neg, scores);
  mprl_loss_kernel<<<BATCH, NNEG, 0, stream>>>(scores, out);
}